// NASErecRAM_55233279426701
// MI455X (gfx1250) — compile-verified
//
#include <hip/hip_runtime.h>

typedef __attribute__((ext_vector_type(2))) float v2f;
typedef __attribute__((ext_vector_type(8))) float v8f;

#define M_ROWS 8192
#define D_COLS 4096
#define SPLITS 32                  // rows per split = 256
#define LDS_STRIDE 68              // padded 16x64 tile row stride (floats)

// ---------------------------------------------------------------------------
// K1: scores[m] = dot(states[m,:], current_state) * (1/sqrt(D))
// One wave32 per 16-row tile, v_wmma_f32_16x16x4_f32, K swept 64 at a time
// through an LDS-staged tile (coalesced global loads, conflict-free LDS reads).
// ---------------------------------------------------------------------------
__global__ __launch_bounds__(64) void k_scores_wmma(
    const float* __restrict__ states, const float* __restrict__ cs,
    float* __restrict__ scores)
{
    __shared__ float lds[2][16 * LDS_STRIDE];
    const int lane = threadIdx.x & 31;
    const int wv   = threadIdx.x >> 5;
    const int row0 = (blockIdx.x * 2 + wv) * 16;
    float* tile = lds[wv];

    const int kbase = (lane >= 16) ? 2 : 0;   // ISA A layout: half-wave K split
    const int mrow  = lane & 15;

    v8f c = {0.f, 0.f, 0.f, 0.f, 0.f, 0.f, 0.f, 0.f};

    for (int k0 = 0; k0 < D_COLS; k0 += 64) {
        // Stage 16x64 f32 tile: each row = 32 lanes * float2 (256B coalesced)
        #pragma unroll
        for (int r = 0; r < 16; ++r) {
            const float2 v = *reinterpret_cast<const float2*>(
                states + (size_t)(row0 + r) * D_COLS + k0 + lane * 2);
            tile[r * LDS_STRIDE + lane * 2 + 0] = v.x;
            tile[r * LDS_STRIDE + lane * 2 + 1] = v.y;
        }
        // Wave-private tile: same-wave DS ordering + compiler s_wait_dscnt.
        #pragma unroll
        for (int kk = 0; kk < 64; kk += 4) {
            v2f a, b;
            a.x = tile[mrow * LDS_STRIDE + kk + kbase + 0];
            a.y = tile[mrow * LDS_STRIDE + kk + kbase + 1];
            // every B column = same current_state chunk -> per-lane broadcast
            b.x = cs[k0 + kk + kbase + 0];
            b.y = cs[k0 + kk + kbase + 1];
            c = __builtin_amdgcn_wmma_f32_16x16x4_f32(
                    false, a, false, b, (short)0, c, false, false);
        }
    }
    // C layout: VGPR v, lanes 0-15 -> M=v; lanes 16-31 -> M=v+8. All N equal.
    if ((lane & 15) == 0) {
        const int rb = (lane >= 16) ? 8 : 0;
        #pragma unroll
        for (int v = 0; v < 8; ++v)
            scores[row0 + rb + v] = c[v] * 0.015625f;   // 1/sqrt(4096)
    }
}

// ---------------------------------------------------------------------------
// Block reduction helpers (wave32)
// ---------------------------------------------------------------------------
__device__ __forceinline__ float block_reduce_sum(float v, float* red)
{
    for (int off = 16; off; off >>= 1) v += __shfl_xor(v, off, 32);
    const int lane = threadIdx.x & 31, w = threadIdx.x >> 5;
    const int nw = blockDim.x >> 5;
    if (lane == 0) red[w] = v;
    __syncthreads();
    if (w == 0) {
        float t = (lane < nw) ? red[lane] : 0.f;
        for (int off = 16; off; off >>= 1) t += __shfl_xor(t, off, 32);
        if (lane == 0) red[0] = t;
    }
    __syncthreads();
    v = red[0];
    __syncthreads();
    return v;
}

__device__ __forceinline__ float block_reduce_max(float v, float* red)
{
    for (int off = 16; off; off >>= 1) v = fmaxf(v, __shfl_xor(v, off, 32));
    const int lane = threadIdx.x & 31, w = threadIdx.x >> 5;
    const int nw = blockDim.x >> 5;
    if (lane == 0) red[w] = v;
    __syncthreads();
    if (w == 0) {
        float t = (lane < nw) ? red[lane] : -3.4e38f;
        for (int off = 16; off; off >>= 1) t = fmaxf(t, __shfl_xor(t, off, 32));
        if (lane == 0) red[0] = t;
    }
    __syncthreads();
    v = red[0];
    __syncthreads();
    return v;
}

// ---------------------------------------------------------------------------
// K2: p[m] = exp(s[m]-max) * weights[m]*exp(-0.01*|t-ts[m]|);
//     inv = 1 / (sum(p) + 1e-9 * sum(exp(s-max)))
// ---------------------------------------------------------------------------
__global__ __launch_bounds__(1024) void k_softmax_prep(
    const float* __restrict__ scores, const float* __restrict__ weights,
    const float* __restrict__ ts, const float* __restrict__ tnew,
    float* __restrict__ p, float* __restrict__ inv_denom)
{
    __shared__ float red[32];
    const int tid = threadIdx.x;

    float lmax = -3.4e38f;
    for (int i = tid; i < M_ROWS; i += 1024) lmax = fmaxf(lmax, scores[i]);
    const float smax = block_reduce_max(lmax, red);

    const float t = tnew[0];
    float sp = 0.f, se = 0.f;
    for (int i = tid; i < M_ROWS; i += 1024) {
        const float e = expf(scores[i] - smax);
        const float w = weights[i] * expf(-0.01f * fabsf(t - ts[i]));
        p[i] = e * w;
        sp += e * w;
        se += e;
    }
    const float SP = block_reduce_sum(sp, red);
    const float SE = block_reduce_sum(se, red);
    if (tid == 0) inv_denom[0] = 1.0f / (SP + 1e-9f * SE);
}

// ---------------------------------------------------------------------------
// K3: partials[split][d] = sum over its 256 rows of p[m]*states[m,d].
// Fully coalesced; deterministic (no float atomics). Prefetch next rows.
// ---------------------------------------------------------------------------
__global__ __launch_bounds__(256) void k_accum(
    const float* __restrict__ states, const float* __restrict__ p,
    float* __restrict__ partials)
{
    const int col   = blockIdx.x * 512 + threadIdx.x * 2;
    const int split = blockIdx.y;
    const int m0    = split * (M_ROWS / SPLITS);
    const int m1    = m0 + (M_ROWS / SPLITS);

    float ax = 0.f, ay = 0.f;
    for (int m = m0; m < m1; ++m) {
        const float pm = p[m];
        const float2 s = *reinterpret_cast<const float2*>(
            states + (size_t)m * D_COLS + col);
        // speculative prefetch (dropped on fault past the end)
        __builtin_prefetch(states + (size_t)(m + 8) * D_COLS + col, 0, 0);
        ax = fmaf(pm, s.x, ax);
        ay = fmaf(pm, s.y, ay);
    }
    float2 out = {ax, ay};
    *reinterpret_cast<float2*>(partials + (size_t)split * D_COLS + col) = out;
}

// ---------------------------------------------------------------------------
// K4: acc[d] = sum_s partials[s][d];  v = 0.95*cs + 0.05*acc*inv;  LayerNorm(v)
// ---------------------------------------------------------------------------
__global__ __launch_bounds__(1024) void k_finalize(
    const float* __restrict__ partials, const float* __restrict__ cs,
    const float* __restrict__ inv_denom, float* __restrict__ out)
{
    __shared__ float red[32];
    const int tid = threadIdx.x;
    const float inv = inv_denom[0];

    float v[4];
    float lsum = 0.f;
    #pragma unroll
    for (int j = 0; j < 4; ++j) {
        const int i = tid + j * 1024;
        float acc = 0.f;
        for (int s = 0; s < SPLITS; ++s)
            acc += partials[(size_t)s * D_COLS + i];
        const float x = 0.95f * cs[i] + (1.0f - 0.95f) * acc * inv;
        v[j] = x;
        lsum += x;
    }
    const float mean = block_reduce_sum(lsum, red) * (1.0f / D_COLS);

    float lvar = 0.f;
    #pragma unroll
    for (int j = 0; j < 4; ++j) {
        const float d = v[j] - mean;
        lvar += d * d;
    }
    const float var = block_reduce_sum(lvar, red) * (1.0f / D_COLS);
    const float rstd = rsqrtf(var + 1e-5f);

    #pragma unroll
    for (int j = 0; j < 4; ++j)
        out[tid + j * 1024] = (v[j] - mean) * rstd;
}

// ---------------------------------------------------------------------------
extern "C" void kernel_launch(void* const* d_in, const int* in_sizes, int n_in,
                              void* d_out, int out_size, void* d_ws, size_t ws_size,
                              hipStream_t stream)
{
    (void)in_sizes; (void)n_in; (void)out_size; (void)ws_size;
    const float* states  = (const float*)d_in[0];   // [M, D]
    const float* weights = (const float*)d_in[1];   // [M]
    const float* ts      = (const float*)d_in[2];   // [M]
    const float* cs      = (const float*)d_in[3];   // [D]
    // d_in[4] = sensed_state: appended to bank, not in output math
    const float* tnew    = (const float*)d_in[5];   // [1]
    float* out = (float*)d_out;                     // [D]

    float* ws       = (float*)d_ws;
    float* scores   = ws;                           // 8192
    float* p        = ws + M_ROWS;                  // 8192
    float* inv      = ws + 2 * M_ROWS;              // 1
    float* partials = ws + 2 * M_ROWS + 64;         // 32*4096 (8B aligned)

    k_scores_wmma<<<dim3(M_ROWS / 32), dim3(64), 0, stream>>>(states, cs, scores);
    k_softmax_prep<<<dim3(1), dim3(1024), 0, stream>>>(scores, weights, ts, tnew,
                                                       p, inv);
    k_accum<<<dim3(D_COLS / 512, SPLITS), dim3(256), 0, stream>>>(states, p,
                                                                  partials);
    k_finalize<<<dim3(1), dim3(1024), 0, stream>>>(partials, cs, inv, out);
}